// GCN_77214922048245
// MI455X (gfx1250) — compile-verified
//
#include <hip/hip_runtime.h>

typedef __attribute__((ext_vector_type(2))) float v2f;
typedef __attribute__((ext_vector_type(8))) float v8f;

#define F 128
#define NROWS 65536  // B*N = 8*8192

// ---------------------------------------------------------------------------
// Zero the output accumulator (float4 per thread).
// ---------------------------------------------------------------------------
__global__ __launch_bounds__(256) void zero_kernel(float4* __restrict__ out) {
    size_t i = (size_t)blockIdx.x * blockDim.x + threadIdx.x;
    out[i] = make_float4(0.f, 0.f, 0.f, 0.f);
}

// ---------------------------------------------------------------------------
// Dense GEMM: H0 = X2 @ W0 and H1 = X2 @ W1 in one pass.
// X2: [NROWS, F] row-major fp32, W: [F, F] row-major fp32.
// One wave computes one 16x16 output tile for BOTH weight matrices using
// V_WMMA_F32_16X16X4_F32: the A fragment is loaded once per k-step and feeds
// two independent WMMAs (better co-execution, half the A traffic).
//
// Fragment layouts per CDNA5 ISA 7.12.2 (fp32, wave32):
//   A 16x4:  lanes 0-15 hold M=lane, (VGPR0,VGPR1)=(K=0,K=1);
//            lanes 16-31 hold M=lane-16, (K=2,K=3)  -> one float2 load.
//   B 4x16:  lane n holds N=n%16; lanes 0-15 (K=0,K=1), lanes 16-31 (K=2,K=3).
//   C/D 16x16: VGPR r: lanes 0-15 -> (M=r, N=lane); lanes 16-31 -> (M=8+r).
// ---------------------------------------------------------------------------
__global__ __launch_bounds__(256) void gemm_wmma(const float* __restrict__ X,
                                                 const float* __restrict__ W0,
                                                 const float* __restrict__ W1,
                                                 float* __restrict__ H) {
    const int wave = threadIdx.x >> 5;   // 8 waves: column tiles 0..7
    const int lane = threadIdx.x & 31;
    const int half = lane >> 4;          // 0: lanes 0-15, 1: lanes 16-31
    const int l16  = lane & 15;
    const int r0   = blockIdx.x * 16;    // row tile
    const int c0   = wave * 16;          // col tile

    // Per-lane base pointers for A / B fragments.
    const float* arow = X  + (size_t)(r0 + l16) * F + 2 * half;    // K offset 2*half
    const float* bp0  = W0 + (size_t)(2 * half) * F + (c0 + l16);  // row (K) stride F
    const float* bp1  = W1 + (size_t)(2 * half) * F + (c0 + l16);

    v8f acc0 = {};
    v8f acc1 = {};
    #pragma unroll
    for (int k = 0; k < F; k += 4) {
        v2f a = *(const v2f*)(arow + k);       // (K=k+2h, K=k+2h+1) for row r0+l16
        v2f b0, b1;
        b0.x = bp0[(size_t)k * F];             // W0[k+2h  ][c0+l16]
        b0.y = bp0[(size_t)k * F + F];         // W0[k+2h+1][c0+l16]
        b1.x = bp1[(size_t)k * F];
        b1.y = bp1[(size_t)k * F + F];
        // 8 args: (neg_a, A, neg_b, B, c_mod, C, reuse_a, reuse_b)
        acc0 = __builtin_amdgcn_wmma_f32_16x16x4_f32(false, a, false, b0,
                                                     (short)0, acc0, false, false);
        acc1 = __builtin_amdgcn_wmma_f32_16x16x4_f32(false, a, false, b1,
                                                     (short)0, acc1, false, false);
    }

    // Store D: VGPR r -> row r0 + r + 8*half, col c0 + l16.
    float* d0 = H + (size_t)(r0 + 8 * half) * F + (c0 + l16);
    float* d1 = d0 + (size_t)NROWS * F;
    #pragma unroll
    for (int r = 0; r < 8; ++r) {
        d0[(size_t)r * F] = acc0[r];
        d1[(size_t)r * F] = acc1[r];
    }
}

// ---------------------------------------------------------------------------
// SpMM scatter, both filters in one launch (gridDim.y selects filter):
//   out[rows[e], :] += vals[e] * H[filt][cols[e], :]
// One wave per edge; each lane owns 4 consecutive features (32*4 = 128).
// H and out are L2-resident (32 MB each vs 192 MB L2), so the random gather
// + fp32 atomic scatter stays on-chip. Edge data is wave-uniform: force it
// into SGPRs so the atomic addressing is scalar-base + lane offset.
// ---------------------------------------------------------------------------
__global__ __launch_bounds__(256) void spmm_scatter(const int* __restrict__ rows0,
                                                    const int* __restrict__ cols0,
                                                    const float* __restrict__ vals0,
                                                    const int* __restrict__ rows1,
                                                    const int* __restrict__ cols1,
                                                    const float* __restrict__ vals1,
                                                    const float* __restrict__ H,
                                                    float* __restrict__ out,
                                                    int E) {
    const int e = blockIdx.x * 8 + (threadIdx.x >> 5);
    if (e >= E) return;
    const int lane = threadIdx.x & 31;
    const int filt = blockIdx.y;

    const int*   rows = filt ? rows1 : rows0;
    const int*   cols = filt ? cols1 : cols0;
    const float* vals = filt ? vals1 : vals0;
    const float* Hf   = H + (size_t)filt * (size_t)NROWS * F;

    // Wave-uniform edge data -> SGPRs.
    const int   row = __builtin_amdgcn_readfirstlane(rows[e]);
    const int   col = __builtin_amdgcn_readfirstlane(cols[e]);
    const float v   = __uint_as_float(
                         __builtin_amdgcn_readfirstlane(__float_as_uint(vals[e])));

    const float4 h = *(const float4*)(Hf + (size_t)col * F + lane * 4);
    float* o = out + (size_t)row * F + lane * 4;

    __hip_atomic_fetch_add(o + 0, v * h.x, __ATOMIC_RELAXED, __HIP_MEMORY_SCOPE_AGENT);
    __hip_atomic_fetch_add(o + 1, v * h.y, __ATOMIC_RELAXED, __HIP_MEMORY_SCOPE_AGENT);
    __hip_atomic_fetch_add(o + 2, v * h.z, __ATOMIC_RELAXED, __HIP_MEMORY_SCOPE_AGENT);
    __hip_atomic_fetch_add(o + 3, v * h.w, __ATOMIC_RELAXED, __HIP_MEMORY_SCOPE_AGENT);
}

// ---------------------------------------------------------------------------
// In-place ReLU (float4 per thread).
// ---------------------------------------------------------------------------
__global__ __launch_bounds__(256) void relu_kernel(float4* __restrict__ out) {
    size_t i = (size_t)blockIdx.x * blockDim.x + threadIdx.x;
    float4 v = out[i];
    v.x = fmaxf(v.x, 0.f);
    v.y = fmaxf(v.y, 0.f);
    v.z = fmaxf(v.z, 0.f);
    v.w = fmaxf(v.w, 0.f);
    out[i] = v;
}

// ---------------------------------------------------------------------------
// Launch sequence: zero -> gemm(H0,H1) -> spmm(both filters) -> relu
// d_ws needs 2 * NROWS * F * 4 = 64 MiB for H0 and H1.
// ---------------------------------------------------------------------------
extern "C" void kernel_launch(void* const* d_in, const int* in_sizes, int n_in,
                              void* d_out, int out_size, void* d_ws, size_t ws_size,
                              hipStream_t stream) {
    const float* X     = (const float*)d_in[0];
    const float* W0    = (const float*)d_in[1];
    const float* W1    = (const float*)d_in[2];
    const int*   rows0 = (const int*)d_in[3];
    const int*   cols0 = (const int*)d_in[4];
    const float* vals0 = (const float*)d_in[5];
    const int*   rows1 = (const int*)d_in[6];
    const int*   cols1 = (const int*)d_in[7];
    const float* vals1 = (const float*)d_in[8];

    float* out = (float*)d_out;
    float* H   = (float*)d_ws;                 // [2][NROWS*F] fp32
    const int E = in_sizes[3];

    const int nvec4 = NROWS * F / 4;           // 2,097,152 float4s

    zero_kernel<<<nvec4 / 256, 256, 0, stream>>>((float4*)out);

    gemm_wmma<<<NROWS / 16, 256, 0, stream>>>(X, W0, W1, H);

    spmm_scatter<<<dim3((E + 7) / 8, 2), 256, 0, stream>>>(
        rows0, cols0, vals0, rows1, cols1, vals1, H, out, E);

    relu_kernel<<<nvec4 / 256, 256, 0, stream>>>((float4*)out);
}